// BindingAffinityModel_28260884807712
// MI455X (gfx1250) — compile-verified
//
#include <hip/hip_runtime.h>
#include <hip/hip_bf16.h>

#define BN_EPS 1e-5f

typedef __attribute__((ext_vector_type(2))) float v2f;
typedef __attribute__((ext_vector_type(8))) float v8f;

// ======================================================================
// WMMA fp32 GEMM:  Y[n,DOUT] = X[n,DIN] @ W[DIN,DOUT] (+bias) (+relu)
// V_WMMA_F32_16X16X4_F32, wave32, one 16x16 C tile per wave.
// DIN must be a multiple of 4 (inputs pre-padded), DOUT multiple of 16,
// n multiple of 16.  grid.x = n/16, block = (32, DOUT/16).
// The 16-row A strip is staged in LDS and shared by all DOUT/16 waves.
// A frag (16x4): lanes 0-15 -> K=k,k+1 ; lanes 16-31 -> K=k+2,k+3 ; M=lane&15
// B frag (4x16): same K split ; N = lane&15
// C (16x16, 8 VGPRs): lanes 0-15 -> M=r ; lanes 16-31 -> M=r+8 ; N = lane&15
// ======================================================================
template <int DIN, int DOUT>
__global__ void gemm_wmma_kernel(const float* __restrict__ X,
                                 const float* __restrict__ W,
                                 const float* __restrict__ bias,
                                 float* __restrict__ Y,
                                 int do_relu)
{
    __shared__ float sX[16 * DIN];

    const int lane    = threadIdx.x;
    const int l       = lane & 15;
    const int half    = lane >> 4;
    const int rowBase = blockIdx.x << 4;
    const int colBase = threadIdx.y << 4;

    // cooperative, fully coalesced strip load (strip rows are contiguous)
    const int tid = threadIdx.y * 32 + lane;
    const int nth = (DOUT / 16) * 32;
#pragma unroll
    for (int i = tid; i < 16 * DIN; i += nth)
        sX[i] = X[(size_t)rowBase * DIN + i];
    __syncthreads();

    v8f acc = {0.f, 0.f, 0.f, 0.f, 0.f, 0.f, 0.f, 0.f};
#pragma unroll
    for (int k = 0; k < DIN; k += 4) {
        const int ka = k + (half << 1);
        v2f a, b;
        a.x = sX[l * DIN + ka];
        a.y = sX[l * DIN + ka + 1];
        b.x = W[(size_t)ka * DOUT + colBase + l];
        b.y = W[(size_t)(ka + 1) * DOUT + colBase + l];
        acc = __builtin_amdgcn_wmma_f32_16x16x4_f32(false, a, false, b,
                                                    (short)0, acc, false, false);
    }

    const float bv = bias ? bias[colBase + l] : 0.f;
#pragma unroll
    for (int r = 0; r < 8; ++r) {
        float v = acc[r] + bv;
        if (do_relu) v = fmaxf(v, 0.f);
        Y[(size_t)(rowBase + (half << 3) + r) * DOUT + colBase + l] = v;
    }
}

// ======================================================================
// Elementwise / scatter helpers
// ======================================================================

// pad [n,din] -> [n,DP] with zero columns (DP multiple of 4)
__global__ void pad_kernel(const float* __restrict__ X, float* __restrict__ Y,
                           int n, int din, int dp)
{
    int i = blockIdx.x * blockDim.x + threadIdx.x;
    if (i >= n * dp) return;
    int r = i / dp, c = i - r * dp;
    Y[i] = (c < din) ? X[(size_t)r * din + c] : 0.f;
}

__global__ void zero_kernel(float* __restrict__ p, int total)
{
    int i = blockIdx.x * blockDim.x + threadIdx.x;
    if (i < total) p[i] = 0.f;
}

// in-degree count via f32 atomics into pre-zeroed buffer
__global__ void deg_count_kernel(const int* __restrict__ dst,
                                 float* __restrict__ deg, int E)
{
    int e = blockIdx.x * blockDim.x + threadIdx.x;
    if (e < E) atomicAdd(&deg[dst[e]], 1.0f);
}

// dinv = rsqrt(deg + 1)   (self loop)
__global__ void deg_fin_kernel(float* __restrict__ deg, int n)
{
    int i = blockIdx.x * blockDim.x + threadIdx.x;
    if (i < n) deg[i] = rsqrtf(deg[i] + 1.0f);
}

// OUT[i,c] = H[i,c] * dinv[i]^2 + bias[c]   (self-loop term + bias)
template <int C>
__global__ void selfinit_kernel(const float* __restrict__ H,
                                const float* __restrict__ dinv,
                                const float* __restrict__ bias,
                                float* __restrict__ OUT, int total)
{
    int i = blockIdx.x * blockDim.x + threadIdx.x;
    if (i >= total) return;
    const int col  = i & (C - 1);
    const int node = i / C;
    const float di = dinv[node];
    OUT[i] = H[i] * di * di + bias[col];
}

// OUT[dst] += H[src] * dinv[src]*dinv[dst]   (float4 per thread, f32 atomics)
template <int C>
__global__ void edge_agg_kernel(const float* __restrict__ H,
                                float* __restrict__ OUT,
                                const int* __restrict__ src,
                                const int* __restrict__ dst,
                                const float* __restrict__ dinv, int E)
{
    constexpr int GPE = C / 4;        // float4 groups per edge
    int idx = blockIdx.x * blockDim.x + threadIdx.x;
    int e = idx / GPE;
    if (e >= E) return;
    int g = (idx - e * GPE) << 2;
    const int s = src[e];
    const int d = dst[e];
    const float w = dinv[s] * dinv[d];
    const float4 h = *reinterpret_cast<const float4*>(H + (size_t)s * C + g);
    float* o = OUT + (size_t)d * C + g;
    atomicAdd(o + 0, h.x * w);
    atomicAdd(o + 1, h.y * w);
    atomicAdd(o + 2, h.z * w);
    atomicAdd(o + 3, h.w * w);
}

// x = relu((x - m) * rsqrt(v + eps) * g + b)
template <int C>
__global__ void bn_relu_kernel(float* __restrict__ X,
                               const float* __restrict__ g,
                               const float* __restrict__ b,
                               const float* __restrict__ m,
                               const float* __restrict__ v, int total)
{
    int i = blockIdx.x * blockDim.x + threadIdx.x;
    if (i >= total) return;
    const int c = i & (C - 1);
    float x = X[i];
    x = (x - m[c]) * rsqrtf(v[c] + BN_EPS) * g[c] + b[c];
    X[i] = fmaxf(x, 0.f);
}

__global__ void relu_kernel(float* __restrict__ X, int total)
{
    int i = blockIdx.x * blockDim.x + threadIdx.x;
    if (i < total) X[i] = fmaxf(X[i], 0.f);
}

// mean-pool: per-graph node counts
__global__ void pool_count_kernel(const int* __restrict__ batch,
                                  float* __restrict__ cnt, int n)
{
    int i = blockIdx.x * blockDim.x + threadIdx.x;
    if (i < n) atomicAdd(&cnt[batch[i]], 1.0f);
}

// scatter node features [n,128] into fused[G,256] at column offset colOff
__global__ void pool_scatter_kernel(const float* __restrict__ X,
                                    const int* __restrict__ batch,
                                    float* __restrict__ fused,
                                    int n, int colOff)
{
    int idx = blockIdx.x * blockDim.x + threadIdx.x;   // n * 32 threads
    int node = idx >> 5;
    if (node >= n) return;
    int g = (idx & 31) << 2;
    const float4 h = *reinterpret_cast<const float4*>(X + (size_t)node * 128 + g);
    float* o = fused + (size_t)batch[node] * 256 + colOff + g;
    atomicAdd(o + 0, h.x);
    atomicAdd(o + 1, h.y);
    atomicAdd(o + 2, h.z);
    atomicAdd(o + 3, h.w);
}

// fused[g, colOff + c] /= max(cnt[g], 1)
__global__ void pool_div_kernel(float* __restrict__ fused,
                                const float* __restrict__ cnt, int colOff)
{
    int i = blockIdx.x * blockDim.x + threadIdx.x;     // 256 * 128 threads
    if (i >= 256 * 128) return;
    int g = i >> 7;
    int c = i & 127;
    fused[(size_t)g * 256 + colOff + c] /= fmaxf(cnt[g], 1.0f);
}

// final layer: out[g] = h3[g,:] @ w[32,1] + b
__global__ void final_kernel(const float* __restrict__ h3,
                             const float* __restrict__ w,
                             const float* __restrict__ b,
                             float* __restrict__ out)
{
    int g = threadIdx.x;   // 256 threads
    float s = 0.f;
#pragma unroll
    for (int k = 0; k < 32; ++k) s += h3[g * 32 + k] * w[k];
    out[g] = s + b[0];
}

// ======================================================================
// Host-side sub-GNN sequence (enqueue only; graph-capture safe)
// ======================================================================
static inline int cdiv(int a, int b) { return (a + b - 1) / b; }

static void run_subgnn(const float* x, int din, int dinPad,
                       const int* edge, int E,
                       const int* batch, int n,
                       const float* w1, const float* b1,
                       const float* bn1g, const float* bn1b,
                       const float* bn1m, const float* bn1v,
                       const float* w2, const float* b2,
                       const float* bn2g, const float* bn2b,
                       const float* bn2m, const float* bn2v,
                       const float* w3, const float* b3,
                       float* bufA, float* bufB, float* dinv, float* padX,
                       float* fused, float* cnt, int colOff,
                       hipStream_t stream)
{
    const int* src = edge;
    const int* dst = edge + E;
    const int T = 256;

    // degrees -> dinv
    zero_kernel<<<cdiv(n, T), T, 0, stream>>>(dinv, n);
    deg_count_kernel<<<cdiv(E, T), T, 0, stream>>>(dst, dinv, E);
    deg_fin_kernel<<<cdiv(n, T), T, 0, stream>>>(dinv, n);

    // pad raw features to K-multiple-of-4, also pads W1 K rows implicitly:
    // extra padX columns are zero, but W rows beyond din are never read
    // because the GEMM uses a zero-padded W? -> No: we pad X with zeros and
    // must pad W too. Instead we zero-pad X and use a W with DIN=dinPad by
    // reading only rows < din; simplest correct approach: pad W into padW.
    // Here: since padX cols >= din are zero, any W row value would be
    // multiplied by zero -- but W rows >= din don't exist in memory.  So we
    // stage a padded W as well (tiny) right after padX in the same buffer.
    pad_kernel<<<cdiv(n * dinPad, T), T, 0, stream>>>(x, padX, n, din, dinPad);
    float* padW = padX + (size_t)n * dinPad;                  // [dinPad, 64]
    pad_kernel<<<cdiv(dinPad * 64, T), T, 0, stream>>>(w1, padW, dinPad == 24 ? 22 : 15, 64, 64);
    // note: pad_kernel with din==dp is a plain copy; here we copy the first
    // `din` rows of w1 and must zero rows din..dinPad-1:
    // rows are handled below by zeroing the tail explicitly.
    zero_kernel<<<1, T, 0, stream>>>(padW + (size_t)din * 64, (dinPad - din) * 64);

    // ---- layer 1: dinPad -> 64, BN1, relu ----
    if (dinPad == 24)
        gemm_wmma_kernel<24, 64><<<dim3(n / 16), dim3(32, 4), 0, stream>>>(padX, padW, nullptr, bufB, 0);
    else
        gemm_wmma_kernel<16, 64><<<dim3(n / 16), dim3(32, 4), 0, stream>>>(padX, padW, nullptr, bufB, 0);
    selfinit_kernel<64><<<cdiv(n * 64, T), T, 0, stream>>>(bufB, dinv, b1, bufA, n * 64);
    edge_agg_kernel<64><<<cdiv(E * 16, T), T, 0, stream>>>(bufB, bufA, src, dst, dinv, E);
    bn_relu_kernel<64><<<cdiv(n * 64, T), T, 0, stream>>>(bufA, bn1g, bn1b, bn1m, bn1v, n * 64);

    // ---- layer 2: 64 -> 64, BN2, relu ----
    gemm_wmma_kernel<64, 64><<<dim3(n / 16), dim3(32, 4), 0, stream>>>(bufA, w2, nullptr, bufB, 0);
    selfinit_kernel<64><<<cdiv(n * 64, T), T, 0, stream>>>(bufB, dinv, b2, bufA, n * 64);
    edge_agg_kernel<64><<<cdiv(E * 16, T), T, 0, stream>>>(bufB, bufA, src, dst, dinv, E);
    bn_relu_kernel<64><<<cdiv(n * 64, T), T, 0, stream>>>(bufA, bn2g, bn2b, bn2m, bn2v, n * 64);

    // ---- layer 3: 64 -> 128, relu ----
    gemm_wmma_kernel<64, 128><<<dim3(n / 16), dim3(32, 8), 0, stream>>>(bufA, w3, nullptr, bufB, 0);
    selfinit_kernel<128><<<cdiv(n * 128, T), T, 0, stream>>>(bufB, dinv, b3, bufA, n * 128);
    edge_agg_kernel<128><<<cdiv(E * 32, T), T, 0, stream>>>(bufB, bufA, src, dst, dinv, E);
    relu_kernel<<<cdiv(n * 128, T), T, 0, stream>>>(bufA, n * 128);

    // ---- mean pool into fused[:, colOff:colOff+128] ----
    zero_kernel<<<1, 256, 0, stream>>>(cnt, 256);
    pool_count_kernel<<<cdiv(n, T), T, 0, stream>>>(batch, cnt, n);
    pool_scatter_kernel<<<cdiv(n * 32, T), T, 0, stream>>>(bufA, batch, fused, n, colOff);
    pool_div_kernel<<<cdiv(256 * 128, T), T, 0, stream>>>(fused, cnt, colOff);
}

extern "C" void kernel_launch(void* const* d_in, const int* in_sizes, int n_in,
                              void* d_out, int out_size, void* d_ws, size_t ws_size,
                              hipStream_t stream)
{
    // ---- inputs (setup_inputs dict order) ----
    const float* px = (const float*)d_in[0];
    const int*   pe = (const int*)  d_in[1];
    const int*   pb = (const int*)  d_in[2];
    const float* mx = (const float*)d_in[3];
    const int*   me = (const int*)  d_in[4];
    const int*   mb = (const int*)  d_in[5];

    const float* p_w1  = (const float*)d_in[6];  const float* p_b1 = (const float*)d_in[7];
    const float* p_w2  = (const float*)d_in[8];  const float* p_b2 = (const float*)d_in[9];
    const float* p_w3  = (const float*)d_in[10]; const float* p_b3 = (const float*)d_in[11];
    const float* p_bn1g = (const float*)d_in[12]; const float* p_bn1b = (const float*)d_in[13];
    const float* p_bn1m = (const float*)d_in[14]; const float* p_bn1v = (const float*)d_in[15];
    const float* p_bn2g = (const float*)d_in[16]; const float* p_bn2b = (const float*)d_in[17];
    const float* p_bn2m = (const float*)d_in[18]; const float* p_bn2v = (const float*)d_in[19];

    const float* m_w1  = (const float*)d_in[20]; const float* m_b1 = (const float*)d_in[21];
    const float* m_w2  = (const float*)d_in[22]; const float* m_b2 = (const float*)d_in[23];
    const float* m_w3  = (const float*)d_in[24]; const float* m_b3 = (const float*)d_in[25];
    const float* m_bn1g = (const float*)d_in[26]; const float* m_bn1b = (const float*)d_in[27];
    const float* m_bn1m = (const float*)d_in[28]; const float* m_bn1v = (const float*)d_in[29];
    const float* m_bn2g = (const float*)d_in[30]; const float* m_bn2b = (const float*)d_in[31];
    const float* m_bn2m = (const float*)d_in[32]; const float* m_bn2v = (const float*)d_in[33];

    const float* f_w1 = (const float*)d_in[34]; const float* f_b1 = (const float*)d_in[35];
    const float* f_w2 = (const float*)d_in[36]; const float* f_b2 = (const float*)d_in[37];
    const float* f_w3 = (const float*)d_in[38]; const float* f_b3 = (const float*)d_in[39];
    const float* f_w4 = (const float*)d_in[40]; const float* f_b4 = (const float*)d_in[41];

    const int NP = in_sizes[2];        // 100000
    const int NM = in_sizes[5];        // 50000
    const int EP = in_sizes[1] / 2;    // 1.6M
    const int EM = in_sizes[4] / 2;    // 0.8M

    // ---- workspace carve (floats); mol reuses protein buffers (NM < NP) ----
    float* ws = (float*)d_ws;
    size_t o = 0;
    float* bufA  = ws + o; o += (size_t)NP * 128;
    float* bufB  = ws + o; o += (size_t)NP * 128;
    float* dinv  = ws + o; o += (size_t)NP;
    float* padX  = ws + o; o += (size_t)NP * 24 + 24 * 64;   // padded X + padded W1
    float* fused = ws + o; o += 256 * 256;
    float* cnt   = ws + o; o += 256;
    float* h1    = ws + o; o += 256 * 128;
    float* h2    = ws + o; o += 256 * 64;
    float* h3    = ws + o; o += 256 * 32;

    // fused accumulator must start at zero every call
    zero_kernel<<<cdiv(256 * 256, 256), 256, 0, stream>>>(fused, 256 * 256);

    // ---- protein sub-GNN -> fused[:, 0:128] ----
    run_subgnn(px, 22, 24, pe, EP, pb, NP,
               p_w1, p_b1, p_bn1g, p_bn1b, p_bn1m, p_bn1v,
               p_w2, p_b2, p_bn2g, p_bn2b, p_bn2m, p_bn2v,
               p_w3, p_b3,
               bufA, bufB, dinv, padX, fused, cnt, 0, stream);

    // ---- mol sub-GNN -> fused[:, 128:256] (reuses buffers) ----
    run_subgnn(mx, 15, 16, me, EM, mb, NM,
               m_w1, m_b1, m_bn1g, m_bn1b, m_bn1m, m_bn1v,
               m_w2, m_b2, m_bn2g, m_bn2b, m_bn2m, m_bn2v,
               m_w3, m_b3,
               bufA, bufB, dinv, padX, fused, cnt, 128, stream);

    // ---- fused MLP head (WMMA f32) ----
    gemm_wmma_kernel<256, 128><<<dim3(16), dim3(32, 8), 0, stream>>>(fused, f_w1, f_b1, h1, 1);
    gemm_wmma_kernel<128, 64><<<dim3(16), dim3(32, 4), 0, stream>>>(h1, f_w2, f_b2, h2, 1);
    gemm_wmma_kernel<64, 32><<<dim3(16), dim3(32, 2), 0, stream>>>(h2, f_w3, f_b3, h3, 1);
    final_kernel<<<1, 256, 0, stream>>>(h3, f_w4, f_b4, (float*)d_out);
}